// BA_Shapes_GCN_26371099198062
// MI455X (gfx1250) — compile-verified
//
#include <hip/hip_runtime.h>
#include <hip/hip_bf16.h>

#define D_H 96

typedef float v2f __attribute__((ext_vector_type(2)));
typedef float v8f __attribute__((ext_vector_type(8)));

// ---------------------------------------------------------------- utilities

__global__ __launch_bounds__(256) void fill_kernel(float* __restrict__ p, float v, int n) {
  int i = blockIdx.x * blockDim.x + threadIdx.x;
  if (i < n) p[i] = v;
}

__global__ __launch_bounds__(256) void degree_kernel(const long long* __restrict__ dst,
                                                     float* __restrict__ deg, int E) {
  int e = blockIdx.x * blockDim.x + threadIdx.x;
  if (e < E) atomicAdd(&deg[(int)dst[e]], 1.0f);
}

__global__ __launch_bounds__(256) void rsqrt_kernel(float* __restrict__ d, int n) {
  int i = blockIdx.x * blockDim.x + threadIdx.x;
  if (i < n) d[i] = rsqrtf(d[i]);
}

// ------------------------------------------------- fp32 WMMA GEMM: N x 96 @ 96 x 96
// One wave computes one 16x16 output tile, accumulating over K=96 with 24
// V_WMMA_F32_16X16X4_F32 issues.
// A frag (16x4 f32): lane half (lane>>4) selects K-pair {0,1} vs {2,3};
//   a.x = A[m][k+2*half], a.y = A[m][k+2*half+1], m = lane&15.
// B frag (4x16 f32): row-striped across lanes within a VGPR; same half rule:
//   b.x = B[k+2*half][n], b.y = B[k+2*half+1][n], n = lane&15.
// C/D (16x16 f32, 8 VGPRs): VGPR r -> row (r + 8*half), col = lane&15.
__global__ __launch_bounds__(128) void gemm96_wmma_kernel(const float* __restrict__ H,
                                                          const float* __restrict__ W,
                                                          float* __restrict__ out, int nrows) {
  const int lane  = threadIdx.x & 31;
  const int gwave = blockIdx.x * (blockDim.x >> 5) + (threadIdx.x >> 5);
  const int tiles_m = (nrows + 15) >> 4;
  const int tile_n = gwave % 6;       // 96 / 16 = 6 column tiles
  const int tile_m = gwave / 6;
  if (tile_m >= tiles_m) return;      // wave-uniform: whole wave exits together

  const int half = lane >> 4;
  const int ln   = lane & 15;
  int row  = tile_m * 16 + ln;
  int rowc = row < nrows ? row : nrows - 1;   // clamp reads: EXEC stays all-ones
  const float* __restrict__ Arow = H + (size_t)rowc * D_H;
  const int ncol = tile_n * 16 + ln;

  v8f acc = {};
#pragma unroll
  for (int k = 0; k < D_H; k += 4) {
    const int kk = k + half * 2;
    v2f a, b;
    a.x = Arow[kk];
    a.y = Arow[kk + 1];
    b.x = W[(size_t)kk * D_H + ncol];
    b.y = W[(size_t)(kk + 1) * D_H + ncol];
    acc = __builtin_amdgcn_wmma_f32_16x16x4_f32(
        /*neg_a=*/false, a, /*neg_b=*/false, b,
        /*c_mod=*/(short)0, acc, /*reuse_a=*/false, /*reuse_b=*/false);
  }

  const int mbase = tile_m * 16 + half * 8;
#pragma unroll
  for (int r = 0; r < 8; ++r) {
    int orow = mbase + r;
    if (orow < nrows) out[(size_t)orow * D_H + ncol] = acc[r];
  }
}

// --------------------------------- self-loop contribution + bias initializes agg
__global__ __launch_bounds__(256) void init_agg_kernel(const float* __restrict__ Hw,
                                                       const float* __restrict__ dinv,
                                                       const float* __restrict__ bias,
                                                       float* __restrict__ out, int n) {
  int i = blockIdx.x * blockDim.x + threadIdx.x;
  if (i < n) {
    int node = i / D_H;
    int j = i - node * D_H;
    float di = dinv[node];
    out[i] = Hw[i] * di * di + bias[j];
  }
}

// ------------------------------------- wave-per-edge normalized scatter-add
__global__ __launch_bounds__(256) void scatter_kernel(const float* __restrict__ Hw,
                                                      const float* __restrict__ dinv,
                                                      const long long* __restrict__ src,
                                                      const long long* __restrict__ dst,
                                                      float* __restrict__ agg, int E) {
  const int e = (blockIdx.x * blockDim.x + threadIdx.x) >> 5;
  if (e >= E) return;
  const int lane = threadIdx.x & 31;
  const int s = (int)src[e];
  const int d = (int)dst[e];
  const float nrm = dinv[s] * dinv[d];
  const float* __restrict__ hs = Hw + (size_t)s * D_H;
  float* __restrict__ od = agg + (size_t)d * D_H;
#pragma unroll
  for (int j = 0; j < 3; ++j) {       // 96 = 32 lanes x 3
    int c = lane + j * 32;
    atomicAdd(&od[c], hs[c] * nrm);   // global_atomic_add_f32, no return
  }
}

__global__ __launch_bounds__(256) void relu_kernel(float* __restrict__ p, int n) {
  int i = blockIdx.x * blockDim.x + threadIdx.x;
  if (i < n) p[i] = fmaxf(p[i], 0.0f);
}

// --------------------------------------------- 96 -> 4 head + log_softmax
__global__ __launch_bounds__(256) void head_kernel(const float* __restrict__ H,
                                                   const float* __restrict__ Wl,
                                                   const float* __restrict__ bl,
                                                   float* __restrict__ out, int N) {
  int i = blockIdx.x * blockDim.x + threadIdx.x;
  if (i >= N) return;
  const float* __restrict__ h = H + (size_t)i * D_H;
  float z0 = bl[0], z1 = bl[1], z2 = bl[2], z3 = bl[3];
#pragma unroll 8
  for (int j = 0; j < D_H; ++j) {
    float hv = h[j];
    z0 = fmaf(hv, Wl[j * 4 + 0], z0);
    z1 = fmaf(hv, Wl[j * 4 + 1], z1);
    z2 = fmaf(hv, Wl[j * 4 + 2], z2);
    z3 = fmaf(hv, Wl[j * 4 + 3], z3);
  }
  float m = fmaxf(fmaxf(z0, z1), fmaxf(z2, z3));
  float s = __expf(z0 - m) + __expf(z1 - m) + __expf(z2 - m) + __expf(z3 - m);
  float lse = m + __logf(s);
  float* o = out + (size_t)i * 4;
  o[0] = z0 - lse; o[1] = z1 - lse; o[2] = z2 - lse; o[3] = z3 - lse;
}

// ---------------------------------------------------------------- launcher

extern "C" void kernel_launch(void* const* d_in, const int* in_sizes, int n_in,
                              void* d_out, int out_size, void* d_ws, size_t ws_size,
                              hipStream_t stream) {
  const float*     x  = (const float*)d_in[0];
  const long long* ei = (const long long*)d_in[1];   // int64 edge_index, shape [2, E]
  const float* Wmat[4] = {(const float*)d_in[2], (const float*)d_in[4],
                          (const float*)d_in[6], (const float*)d_in[8]};
  const float* bias[4] = {(const float*)d_in[3], (const float*)d_in[5],
                          (const float*)d_in[7], (const float*)d_in[9]};
  const float* Wl = (const float*)d_in[10];
  const float* bl = (const float*)d_in[11];

  const int N = in_sizes[0] / D_H;
  const int E = in_sizes[1] / 2;
  const long long* src = ei;
  const long long* dst = ei + E;

  float* ws   = (float*)d_ws;
  float* dinv = ws;                                  // N
  float* Hw   = dinv + N;                            // N*96 (transformed features)
  float* bufA = Hw + (size_t)N * D_H;                // N*96 (layer ping)
  float* bufB = bufA + (size_t)N * D_H;              // N*96 (layer pong)

  const int T = 256;
  const int nelem = N * D_H;

  // degrees -> dinv
  fill_kernel<<<(N + T - 1) / T, T, 0, stream>>>(dinv, 1.0f, N);   // self-loop
  degree_kernel<<<(E + T - 1) / T, T, 0, stream>>>(dst, dinv, E);
  rsqrt_kernel<<<(N + T - 1) / T, T, 0, stream>>>(dinv, N);

  const int gemm_waves  = ((N + 15) / 16) * 6;
  const int gemm_blocks = (gemm_waves + 3) / 4;                    // 4 waves / block
  const int scat_blocks = (int)(((long long)E * 32 + T - 1) / T);  // wave per edge

  const float* Hin = x;
  float* bufs[2] = {bufA, bufB};
  for (int l = 0; l < 4; ++l) {
    float* Hout = bufs[l & 1];
    gemm96_wmma_kernel<<<gemm_blocks, 128, 0, stream>>>(Hin, Wmat[l], Hw, N);
    init_agg_kernel<<<(nelem + T - 1) / T, T, 0, stream>>>(Hw, dinv, bias[l], Hout, nelem);
    scatter_kernel<<<scat_blocks, T, 0, stream>>>(Hw, dinv, src, dst, Hout, E);
    relu_kernel<<<(nelem + T - 1) / T, T, 0, stream>>>(Hout, nelem);
    Hin = Hout;
  }

  head_kernel<<<(N + T - 1) / T, T, 0, stream>>>(Hin, Wl, bl, (float*)d_out, N);
}